// DiffusionBlock_70609262346828
// MI455X (gfx1250) — compile-verified
//
#include <hip/hip_runtime.h>
#include <stdint.h>

// ---------------------------------------------------------------------------
// Fused 20-step 5-point diffusion (reflect BC) for MI455X / gfx1250.
//
// Temporal blocking: each workgroup stages a 104x104 fp32 tile (64x64 output
// + halo 20 = #steps) into LDS via CDNA5 async global->LDS DMA (ASYNCcnt,
// b128 fast path for interior tiles, mirror-indexed b32 path for border
// tiles), runs all 20 Jacobi steps in LDS ping-pong buffers with a vertical
// register pipeline (3 LDS loads + 1 store per point) and a validity-cone-
// shrunk row band, then writes the 64x64 interior with b128 stores.
// HBM: ~244 MiB total vs 2.68 GB for 20 stepwise passes (~11 us vs ~115 us
// at 23.3 TB/s).
// ---------------------------------------------------------------------------

namespace {
constexpr int IMG_H = 1024;
constexpr int IMG_W = 1024;
constexpr int BATCH = 16;
constexpr int STEPS = 20;
constexpr int HALO  = STEPS;               // fusion halo
constexpr int TILE  = 64;                  // output tile (divides 1024)
constexpr int EXT   = TILE + 2 * HALO;     // 104
constexpr int NT    = 256;                 // 8 wave32s
constexpr size_t SMEM_BYTES = (size_t)2 * EXT * EXT * sizeof(float); // 86.5 KB
}

// Group-segment byte offset of an LDS pointer: generic LDS addresses on
// gfx1250 carry the aperture in the high bits, wave-relative offset in [31:0].
__device__ __forceinline__ unsigned lds_offset(const void* p) {
  return (unsigned)(uintptr_t)p;
}

// ASYNCcnt-tracked async DMA: copies global memory at base + goff to the LDS byte offset doff
__device__ __forceinline__ void async_lds_b32(unsigned doff, unsigned goff,
                                              uint64_t base) {
  asm volatile("global_load_async_to_lds_b32 %0, %1, %2 offset:0"
               :: "v"(doff), "v"(goff), "s"(base) : "memory");
}
__device__ __forceinline__ void async_lds_b128(unsigned doff, unsigned goff,
                                               uint64_t base) {
  asm volatile("global_load_async_to_lds_b128 %0, %1, %2 offset:0"
               :: "v"(doff), "v"(goff), "s"(base) : "memory");
}

__global__ __launch_bounds__(NT)
void diffusion_fused20(const float* __restrict__ x,
                       const float* __restrict__ w3x3,
                       float* __restrict__ out) {
  extern __shared__ float smem[];
  float* buf0 = smem;
  float* buf1 = smem + EXT * EXT;

  const int tid  = threadIdx.x;
  const int lane = tid & 31;
  const int wv   = tid >> 5;
  const int tx   = blockIdx.x;
  const int ty   = blockIdx.y;
  const int b    = blockIdx.z;

  const float* img  = x   + (size_t)b * IMG_H * IMG_W;
  float*       oimg = out + (size_t)b * IMG_H * IMG_W;
  const int gx0 = tx * TILE - HALO;
  const int gy0 = ty * TILE - HALO;
  const uint64_t base = (uint64_t)(uintptr_t)img;
  const unsigned lds0 = lds_offset(buf0);

  // ---- Stage 104x104 tile into LDS (async DMA) ----
  const bool interior = (tx > 0) & (tx < IMG_W / TILE - 1) &
                        (ty > 0) & (ty < IMG_H / TILE - 1);
  if (interior) {
    // Fully in-bounds; 16B-aligned on both global and LDS side.
    for (int p4 = tid; p4 < EXT * EXT / 4; p4 += NT) {
      const int er = p4 / (EXT / 4);
      const int e4 = (p4 - er * (EXT / 4)) * 4;
      const unsigned goff = (unsigned)(((gy0 + er) * IMG_W + gx0 + e4) * 4);
      async_lds_b128(lds0 + (unsigned)((er * EXT + e4) * 4), goff, base);
    }
  } else {
    // Mirror (reflect) indexing; exact for fused steps since the mirror
    // extension is invariant under the symmetric stencil.
    for (int p = tid; p < EXT * EXT; p += NT) {
      const int er = p / EXT;
      const int ec = p - er * EXT;
      int gr = gy0 + er;
      gr = (gr < 0) ? -gr : ((gr > IMG_H - 1) ? 2 * (IMG_H - 1) - gr : gr);
      int gc = gx0 + ec;
      gc = (gc < 0) ? -gc : ((gc > IMG_W - 1) ? 2 * (IMG_W - 1) - gc : gc);
      const unsigned goff = (unsigned)((gr * IMG_W + gc) * 4);
      async_lds_b32(lds0 + (unsigned)(p * 4), goff, base);
    }
  }
  // Zero the ping buffer's outer ring: it is never computed and only feeds
  // throwaway fringe points, but this keeps every LDS read deterministic.
  for (int p = tid; p < EXT; p += NT) {
    buf1[p] = 0.f;
    buf1[(EXT - 1) * EXT + p] = 0.f;
  }
  for (int p = tid; p < EXT - 2; p += NT) {
    buf1[(p + 1) * EXT] = 0.f;
    buf1[(p + 1) * EXT + EXT - 1] = 0.f;
  }
  asm volatile("s_wait_asynccnt 0" ::: "memory");  // drain this wave's DMA
  __syncthreads();

  // Stencil taps from the 3x3 OIHW weight (corners are zero by construction).
  const float wN = w3x3[1];
  const float wW = w3x3[3];
  const float wC = w3x3[4];
  const float wE = w3x3[5];
  const float wS = w3x3[7];

  float* cur = buf0;
  float* nxt = buf1;

#pragma unroll 1
  for (int s = 0; s < STEPS; ++s) {
    // Validity cone: only ext rows [s+1, EXT-2-s] can still matter.
    const int rlo = 1 + s;
    const int rhi = EXT - 2 - s;                 // inclusive
    const int nrows = rhi - rlo + 1;             // 102 - 2s (even, >= 64)
    const int per = (nrows + 7) >> 3;            // rows per wave band
    const int r0 = rlo + wv * per;
    int r1 = r0 + per;
    if (r1 > rhi + 1) r1 = rhi + 1;

#pragma unroll
    for (int cb = 0; cb < 4; ++cb) {
      const int c = 1 + lane + cb * 32;          // lane-consecutive columns
      if (c <= EXT - 2 && r0 < r1) {
        // Vertical register pipeline: N and C taps live in registers.
        float vN = cur[(r0 - 1) * EXT + c];
        float vC = cur[r0 * EXT + c];
#pragma unroll 1
        for (int r = r0; r < r1; ++r) {
          const float vS = cur[(r + 1) * EXT + c];
          const float vW = cur[r * EXT + c - 1];
          const float vE = cur[r * EXT + c + 1];
          nxt[r * EXT + c] = wC * vC + wN * vN + wS * vS + wW * vW + wE * vE;
          vN = vC;
          vC = vS;
        }
      }
    }
    __syncthreads();
    float* t = cur; cur = nxt; nxt = t;
  }

  // ---- Write the valid 64x64 interior (16B-aligned b128 LDS reads/stores) --
  for (int p = tid; p < TILE * (TILE / 4); p += NT) {
    const int r  = p / (TILE / 4);
    const int c4 = (p - r * (TILE / 4)) * 4;
    const float4 v = *(const float4*)&cur[(HALO + r) * EXT + HALO + c4];
    *(float4*)&oimg[(size_t)(ty * TILE + r) * IMG_W + tx * TILE + c4] = v;
  }
}

extern "C" void kernel_launch(void* const* d_in, const int* in_sizes, int n_in,
                              void* d_out, int out_size, void* d_ws, size_t ws_size,
                              hipStream_t stream) {
  (void)in_sizes; (void)n_in; (void)out_size; (void)d_ws; (void)ws_size;

  const float* x = (const float*)d_in[0];   // (16,1,1024,1024) fp32
  const float* w = (const float*)d_in[1];   // (1,1,3,3) fp32
  // d_in[2] = time_steps (20): fused at compile time as STEPS/HALO.
  float* out = (float*)d_out;

  static_assert(SMEM_BYTES == 86528, "LDS budget");
  (void)hipFuncSetAttribute((const void*)diffusion_fused20,
                            hipFuncAttributeMaxDynamicSharedMemorySize,
                            (int)SMEM_BYTES);

  dim3 grid(IMG_W / TILE, IMG_H / TILE, BATCH);   // 16 x 16 x 16
  diffusion_fused20<<<grid, NT, SMEM_BYTES, stream>>>(x, w, out);
}